// ExpertChoiceRouter_33809982554647
// MI455X (gfx1250) — compile-verified
//
#include <hip/hip_runtime.h>
#include <hip/hip_bf16.h>
#include <math.h>

// Problem sizes (fixed by the reference)
#define N_TOK 16384   // B*T
#define D_DIM 2048
#define E_EXP 64
#define CAP   512

typedef __attribute__((ext_vector_type(2))) float v2f;
typedef __attribute__((ext_vector_type(8))) float v8f;

// ---------------------------------------------------------------------------
// Kernel 1: router GEMM (fp32 WMMA 16x16x4) + fused softmax.
// One wave (32 lanes) computes a 16-token x 64-expert logits tile.
// Block = 256 threads = 8 waves; grid = 128 blocks -> 1024 tiles = 16384 tokens.
// ---------------------------------------------------------------------------
__global__ __launch_bounds__(256)
void router_gemm_softmax(const float* __restrict__ x,
                         const float* __restrict__ W,
                         float* __restrict__ probs_out,  // [N, E]
                         float* __restrict__ probs_t)    // [E, N] (workspace)
{
    // 8 waves * 16 tokens * 65-float padded rows (65 kills LDS bank conflicts)
    __shared__ float lds[8][16 * 65];

    const int tid  = threadIdx.x;
    const int wave = tid >> 5;
    const int lane = tid & 31;
    const int m    = lane & 15;   // token row within tile / expert col within tile
    const int h    = lane >> 4;   // selects K pair {0,1} vs {2,3}
    const int tile = blockIdx.x * 8 + wave;
    const int token0 = tile * 16;

    const float* __restrict__ xrow = x + (size_t)(token0 + m) * D_DIM;
    const float* __restrict__ w0 = W + (size_t)(m +  0) * D_DIM;
    const float* __restrict__ w1 = W + (size_t)(m + 16) * D_DIM;
    const float* __restrict__ w2 = W + (size_t)(m + 32) * D_DIM;
    const float* __restrict__ w3 = W + (size_t)(m + 48) * D_DIM;

    v8f c0 = {}, c1 = {}, c2 = {}, c3 = {};

    #pragma unroll 4
    for (int k0 = 0; k0 < D_DIM; k0 += 4) {
        const int k = k0 + 2 * h;
        // A (16x4 f32): lane(m,h) holds x[m][k], x[m][k+1]
        v2f a = *(const v2f*)(xrow + k);
        // B (4x16 f32): lane(n=m,h) holds W[e][k], W[e][k+1] for its expert column
        v2f b0 = *(const v2f*)(w0 + k);
        v2f b1 = *(const v2f*)(w1 + k);
        v2f b2 = *(const v2f*)(w2 + k);
        v2f b3 = *(const v2f*)(w3 + k);
        c0 = __builtin_amdgcn_wmma_f32_16x16x4_f32(false, a, false, b0, (short)0, c0, false, false);
        c1 = __builtin_amdgcn_wmma_f32_16x16x4_f32(false, a, false, b1, (short)0, c1, false, false);
        c2 = __builtin_amdgcn_wmma_f32_16x16x4_f32(false, a, false, b2, (short)0, c2, false, false);
        c3 = __builtin_amdgcn_wmma_f32_16x16x4_f32(false, a, false, b3, (short)0, c3, false, false);
    }

    // Spill C tiles to LDS. C layout: lane L holds N = L&15; VGPR v holds M = v + 8*(L>>4).
    float* wl = &lds[wave][0];
    #pragma unroll
    for (int v = 0; v < 8; ++v) {
        const int mm = v + 8 * h;
        wl[mm * 65 +  0 + m] = c0[v];
        wl[mm * 65 + 16 + m] = c1[v];
        wl[mm * 65 + 32 + m] = c2[v];
        wl[mm * 65 + 48 + m] = c3[v];
    }
    __syncthreads();

    // Softmax over 64 experts; lanes 0..15 each own one token of the tile.
    if (lane < 16) {
        float* row = wl + lane * 65;
        float mx = -INFINITY;
        #pragma unroll
        for (int e = 0; e < E_EXP; ++e) mx = fmaxf(mx, row[e]);
        float sum = 0.0f;
        #pragma unroll
        for (int e = 0; e < E_EXP; ++e) {
            float p = __expf(row[e] - mx);
            row[e] = p;
            sum += p;
        }
        const float inv = 1.0f / sum;
        const int tok = token0 + lane;
        #pragma unroll
        for (int e = 0; e < E_EXP; ++e) {
            const float p = row[e] * inv;
            probs_out[(size_t)tok * E_EXP + e] = p;       // [N, E]
            probs_t[(size_t)e * N_TOK + tok]  = p;        // [E, N] coalesced for phase 2
        }
    }
}

// ---------------------------------------------------------------------------
// Kernel 2: per-expert exact top-512 (radix select on float bit patterns,
// which are order-preserving for positive softmax probs) + bitonic sort
// (val desc, idx asc) + normalization.  One 512-thread block per expert.
// ---------------------------------------------------------------------------
__global__ __launch_bounds__(512)
void expert_topk(const float* __restrict__ probs_t,  // [E, N]
                 int* __restrict__ out_idx,          // [E, CAP]
                 float* __restrict__ out_w)          // [E, CAP]
{
    __shared__ unsigned hist[256];
    __shared__ unsigned sh_prefix, sh_k;
    __shared__ unsigned cnt_g, cnt_t;
    __shared__ float s_val[CAP];
    __shared__ int   s_idx[CAP];
    __shared__ float s_sum[CAP];

    const int e   = blockIdx.x;
    const int tid = threadIdx.x;
    const float* __restrict__ row = probs_t + (size_t)e * N_TOK;

    if (tid == 0) { sh_prefix = 0u; sh_k = CAP; }
    __syncthreads();

    unsigned prefmask = 0u;
    for (int pass = 0; pass < 4; ++pass) {
        const int shift = 24 - 8 * pass;
        if (tid < 256) hist[tid] = 0u;
        __syncthreads();
        const unsigned prefix = sh_prefix;
        for (int i = tid; i < N_TOK; i += 512) {
            const unsigned key = __float_as_uint(row[i]);
            if ((key & prefmask) == prefix)
                atomicAdd(&hist[(key >> shift) & 255u], 1u);
        }
        __syncthreads();
        if (tid == 0) {
            const unsigned k = sh_k;
            unsigned cum = 0;
            int b = 255;
            for (; b > 0; --b) {
                const unsigned c = hist[b];
                if (cum + c >= k) break;
                cum += c;
            }
            sh_prefix = prefix | ((unsigned)b << shift);
            sh_k = k - cum;  // rank remaining within chosen bin
        }
        __syncthreads();
        prefmask |= (0xFFu << shift);
    }

    const unsigned thr = sh_prefix;   // exact bit pattern of the 512th-largest value
    const unsigned T   = sh_k;        // number of ties (== thr) to keep
    const unsigned G   = CAP - T;     // count of strictly-greater values

    if (tid == 0) { cnt_g = 0u; cnt_t = 0u; }
    __syncthreads();

    // Compaction into LDS: greater-than elements at [0,G), ties at [G, G+T).
    for (int i = tid; i < N_TOK; i += 512) {
        const unsigned key = __float_as_uint(row[i]);
        if (key > thr) {
            const unsigned p = atomicAdd(&cnt_g, 1u);
            s_val[p] = __uint_as_float(key);
            s_idx[p] = i;
        } else if (key == thr) {
            const unsigned t = atomicAdd(&cnt_t, 1u);
            if (t < T) {
                s_val[G + t] = __uint_as_float(key);
                s_idx[G + t] = i;
            }
        }
    }
    __syncthreads();

    // Bitonic sort of 512 (val, idx): val descending, idx ascending on ties.
    for (int k2 = 2; k2 <= CAP; k2 <<= 1) {
        for (int j = k2 >> 1; j > 0; j >>= 1) {
            const int ixj = tid ^ j;
            if (ixj > tid) {
                const bool desc = ((tid & k2) == 0);
                const float v0 = s_val[tid], v1 = s_val[ixj];
                const int   i0 = s_idx[tid], i1 = s_idx[ixj];
                const bool firstBefore = (v0 > v1) || (v0 == v1 && i0 < i1);
                const bool doSwap = desc ? !firstBefore : firstBefore;
                if (doSwap) {
                    s_val[tid] = v1; s_val[ixj] = v0;
                    s_idx[tid] = i1; s_idx[ixj] = i0;
                }
            }
            __syncthreads();
        }
    }

    // Sum of the 512 selected weights -> normalize.
    s_sum[tid] = s_val[tid];
    __syncthreads();
    for (int s = 256; s > 0; s >>= 1) {
        if (tid < s) s_sum[tid] += s_sum[tid + s];
        __syncthreads();
    }
    const float denom = fmaxf(s_sum[0], 1e-8f);

    out_idx[e * CAP + tid] = s_idx[tid];
    out_w[e * CAP + tid]   = s_val[tid] / denom;
}

// ---------------------------------------------------------------------------
// Launch.  d_out layout (flat, reference return order):
//   [0, E*CAP)            expert_indices (int32 bits)
//   [E*CAP, 2*E*CAP)      expert_weights (f32)
//   [2*E*CAP, +N*E)       routing_probs (f32)
// d_ws: probs transposed [E, N] (4 MB).
// ---------------------------------------------------------------------------
extern "C" void kernel_launch(void* const* d_in, const int* in_sizes, int n_in,
                              void* d_out, int out_size, void* d_ws, size_t ws_size,
                              hipStream_t stream) {
    const float* x = (const float*)d_in[0];   // [B,T,D] fp32
    const float* W = (const float*)d_in[1];   // [E,D]   fp32

    float* probs_t   = (float*)d_ws;                          // [E, N]
    int*   out_idx   = (int*)d_out;                           // [E, CAP]
    float* out_w     = (float*)d_out + (size_t)E_EXP * CAP;   // [E, CAP]
    float* probs_out = (float*)d_out + (size_t)2 * E_EXP * CAP; // [N, E]

    // Phase 1: 1024 token tiles, 8 waves/block -> 128 blocks.
    router_gemm_softmax<<<dim3(N_TOK / 16 / 8), dim3(256), 0, stream>>>(
        x, W, probs_out, probs_t);

    // Phase 2: one block per expert.
    expert_topk<<<dim3(E_EXP), dim3(512), 0, stream>>>(probs_t, out_idx, out_w);
}